// DotAttention_32306744000884
// MI455X (gfx1250) — compile-verified
//
#include <hip/hip_runtime.h>
#include <hip/hip_bf16.h>
#include <math.h>

typedef __attribute__((ext_vector_type(2))) float v2f;
typedef __attribute__((ext_vector_type(8))) float v8f;

#define NB 64
#define NS 4096
#define NH 1024
#define ROWS 128          // s-rows per scores block
#define KC 32             // h-chunk per LDS tile (double buffered)
#define NCHUNK (NH / KC)  // 32 chunks
#define LDSTR 36          // padded row stride (floats): 144B, 16B aligned, conflict-free

// ---------------------------------------------------------------------------
// Kernel 1: scores[b,s] = dot(enc[b,s,:], dec[b,:]) via V_WMMA_F32_16X16X4_F32.
// enc tiles staged global->LDS with GLOBAL_LOAD_ASYNC_TO_LDS_B128 (ASYNCcnt),
// double buffered so copy(chunk c+1) overlaps WMMA(chunk c).
// A = 16 enc rows x 4 k (LDS), B = dec[k] broadcast to all 16 columns
// => every column of C holds the 16 scores.
// ---------------------------------------------------------------------------
__global__ __launch_bounds__(256) void da_scores_kernel(
    const float* __restrict__ enc, const float* __restrict__ dec,
    float* __restrict__ scores) {
  __shared__ float tileA[2][ROWS * LDSTR];  // 2 x 18 KB
  __shared__ float decs[NH];                // 4 KB

  const int b    = blockIdx.y;
  const int s0   = blockIdx.x * ROWS;
  const int t    = threadIdx.x;
  const int lane = t & 31;
  const int wave = t >> 5;         // 0..7, each wave owns 16 s-rows
  const int hi   = lane >> 4;      // half-wave: K pair 0/1 vs 2/3
  const int mrow = lane & 15;      // A-matrix row within wave tile

  for (int i = t; i < NH; i += 256) decs[i] = dec[(size_t)b * NH + i];

  const float* encB = enc + (size_t)b * NS * NH;

  // LDS byte addresses: flat pointers to LDS carry the LDS offset in addr[31:0]
  const unsigned ldsBase0 = (unsigned)(uintptr_t)(&tileA[0][0]);
  const unsigned ldsBase1 = (unsigned)(uintptr_t)(&tileA[1][0]);

  // This thread's fixed slice of each [ROWS x KC] tile: 4 x b128 per tile.
  // idx = p*256 + t ; row = idx/8 ; c4 = idx%8  (8 float4 per row)
  int rr[4], cc4[4];
  #pragma unroll
  for (int p = 0; p < 4; ++p) {
    int idx = p * 256 + t;
    rr[p]  = idx >> 3;
    cc4[p] = idx & 7;
  }

  auto issue_tile = [&](int bufsel, int h0) {
    const unsigned lbase = bufsel ? ldsBase1 : ldsBase0;
    #pragma unroll
    for (int p = 0; p < 4; ++p) {
      const float* g = encB + (size_t)(s0 + rr[p]) * NH + h0 + cc4[p] * 4;
      unsigned l = lbase + (unsigned)((rr[p] * LDSTR + cc4[p] * 4) * 4);
      asm volatile("global_load_async_to_lds_b128 %0, %1, off"
                   :: "v"(l), "v"(g) : "memory");
    }
  };

  v8f acc = {};

  issue_tile(0, 0);  // prologue: tile for chunk 0

  for (int c = 0; c < NCHUNK; ++c) {
    const int cur = c & 1;
    if (c + 1 < NCHUNK) {
      issue_tile(cur ^ 1, (c + 1) * KC);                 // prefetch next tile
      asm volatile("s_wait_asynccnt 0x4" ::: "memory");  // current tile done
    } else {
      asm volatile("s_wait_asynccnt 0x0" ::: "memory");
    }
    __syncthreads();  // all waves' portions of chunk c are in LDS

    const int h0 = c * KC;
    const float* rowBase = &tileA[cur][(wave * 16 + mrow) * LDSTR];
    #pragma unroll
    for (int k = 0; k < KC; k += 4) {
      v2f a, bb;
      a.x  = rowBase[k + 2 * hi];
      a.y  = rowBase[k + 2 * hi + 1];
      bb.x = decs[h0 + k + 2 * hi];        // broadcast: all N columns identical
      bb.y = decs[h0 + k + 2 * hi + 1];
      acc = __builtin_amdgcn_wmma_f32_16x16x4_f32(
          /*neg_a=*/false, a, /*neg_b=*/false, bb,
          /*c_mod=*/(short)0, acc, /*reuse_a=*/false, /*reuse_b=*/false);
    }
    __syncthreads();  // all waves done reading buf[cur] before it is re-filled
  }

  // C layout: VGPR i on lanes 0-15 -> M=i; lanes 16-31 -> M=8+i (cols all equal)
  if (lane == 0 || lane == 16) {
    const int mbase = (lane == 16) ? 8 : 0;
    float* dst = scores + (size_t)b * NS + s0 + wave * 16 + mbase;
    #pragma unroll
    for (int i = 0; i < 8; ++i) dst[i] = acc[i];
  }
}

// ---------------------------------------------------------------------------
// Kernel 2: softmax over S per batch; writes weights to d_out weights region
// and copies dec into out[:, H:2H].
// ---------------------------------------------------------------------------
__global__ __launch_bounds__(256) void da_softmax_kernel(
    const float* __restrict__ scores, const float* __restrict__ dec,
    float* __restrict__ outp) {
  __shared__ float red[256];
  const int b = blockIdx.x;
  const int t = threadIdx.x;
  const float* sc = scores + (size_t)b * NS;

  float m = -INFINITY;
  for (int i = t; i < NS; i += 256) m = fmaxf(m, sc[i]);
  red[t] = m; __syncthreads();
  for (int o = 128; o > 0; o >>= 1) {
    if (t < o) red[t] = fmaxf(red[t], red[t + o]);
    __syncthreads();
  }
  m = red[0]; __syncthreads();

  float sum = 0.f;
  for (int i = t; i < NS; i += 256) sum += __expf(sc[i] - m);
  red[t] = sum; __syncthreads();
  for (int o = 128; o > 0; o >>= 1) {
    if (t < o) red[t] += red[t + o];
    __syncthreads();
  }
  const float inv = 1.0f / red[0];

  float* wout = outp + (size_t)NB * 2 * NH + (size_t)b * NS;  // weights region
  for (int i = t; i < NS; i += 256) wout[i] = __expf(sc[i] - m) * inv;

  float* o2 = outp + (size_t)b * 2 * NH + NH;                 // out[:, H:2H] = dec
  for (int i = t; i < NH; i += 256) o2[i] = dec[(size_t)b * NH + i];
}

// ---------------------------------------------------------------------------
// Kernel 3: context[b,h] = sum_s w[b,s]*enc[b,s,h] via WMMA.
// A = w[s+k] broadcast across rows, B[k][n] = enc[s+k, h0+n] (16 contiguous
// floats per k-row per half-wave). Every C row equals the context chunk.
// ---------------------------------------------------------------------------
__global__ __launch_bounds__(128) void da_context_kernel(
    const float* __restrict__ enc, const float* __restrict__ weights,
    float* __restrict__ outp) {
  __shared__ float wsh[NS];  // 16 KB
  const int b    = blockIdx.y;
  const int t    = threadIdx.x;
  const int lane = t & 31;
  const int wave = t >> 5;      // 0..3, each wave owns 16 h-columns
  const int hi   = lane >> 4;
  const int n    = lane & 15;

  for (int i = t; i < NS; i += 128) wsh[i] = weights[(size_t)b * NS + i];
  __syncthreads();

  const int h0 = blockIdx.x * 64 + wave * 16;
  const float* encB = enc + (size_t)b * NS * NH;
  v8f acc = {};

  #pragma unroll 4
  for (int s = 0; s < NS; s += 4) {
    v2f a, bb;
    a.x  = wsh[s + 2 * hi];                               // A rows all identical
    a.y  = wsh[s + 2 * hi + 1];
    bb.x = encB[(size_t)(s + 2 * hi) * NH + h0 + n];      // B[k][n]
    bb.y = encB[(size_t)(s + 2 * hi + 1) * NH + h0 + n];
    acc = __builtin_amdgcn_wmma_f32_16x16x4_f32(
        false, a, false, bb, (short)0, acc, false, false);
  }

  // Every row of C is identical; lane n (lanes 0-15) holds column n in acc[0].
  if (lane < 16) outp[(size_t)b * 2 * NH + h0 + n] = acc[0];
}

extern "C" void kernel_launch(void* const* d_in, const int* in_sizes, int n_in,
                              void* d_out, int out_size, void* d_ws, size_t ws_size,
                              hipStream_t stream) {
  const float* enc = (const float*)d_in[0];   // [64, 4096, 1024] f32
  const float* dec = (const float*)d_in[1];   // [64, 1024] f32
  float* outp   = (float*)d_out;              // [64*2048] out ++ [64*4096] weights
  float* scores = (float*)d_ws;               // needs 64*4096*4 = 1 MiB scratch

  da_scores_kernel<<<dim3(NS / ROWS, NB), 256, 0, stream>>>(enc, dec, scores);
  da_softmax_kernel<<<NB, 256, 0, stream>>>(scores, dec, outp);
  const float* wts = outp + (size_t)NB * 2 * NH;  // weights live in d_out
  da_context_kernel<<<dim3(NH / 64, NB), 128, 0, stream>>>(enc, wts, outp);
}